// Loss_28492813041984
// MI455X (gfx1250) — compile-verified
//
#include <hip/hip_runtime.h>
#include <hip/hip_bf16.h>
#include <math.h>

#define BATCH 8
#define NPTS 4096
#define NPOINT 204
#define NSMAX 49
#define NPCT 5

typedef float v2f __attribute__((ext_vector_type(2)));
typedef float v8f __attribute__((ext_vector_type(8)));

__constant__ float c_R2[NPCT] = {0.004f, 0.006f, 0.008f, 0.01f, 0.012f};
__constant__ int   c_NS[NPCT] = {16, 24, 32, 40, 49};
// (p*100)^2 / (8*204*5)
__constant__ float c_W[NPCT]  = {1.9607843e-5f, 4.4117647e-5f, 7.8431373e-5f,
                                 1.2254902e-4f, 1.7647059e-4f};

__global__ void zero_out_kernel(float* out) {
    if (threadIdx.x == 0) out[0] = 0.f;
}

// |p|^2 for every point
__global__ void rr_kernel(const float* __restrict__ pcd, float* __restrict__ rr) {
    int i = blockIdx.x * blockDim.x + threadIdx.x;
    if (i < BATCH * NPTS) {
        float x = pcd[i*3], y = pcd[i*3+1], z = pcd[i*3+2];
        rr[i] = x*x + y*y + z*z;
    }
}

// Farthest point sampling: one block (256 thr) per batch, sequential argmax loop.
__global__ void fps_kernel(const float* __restrict__ pcd, int* __restrict__ fps) {
    int b = blockIdx.x;
    int tid = threadIdx.x;
    const float* P = pcd + b * NPTS * 3;
    float px[16], py[16], pz[16], md[16];
    for (int t = 0; t < 16; ++t) {
        int i = tid + t * 256;
        px[t] = P[i*3]; py[t] = P[i*3+1]; pz[t] = P[i*3+2];
        md[t] = 1e10f;
    }
    __shared__ float s_bv[256];
    __shared__ int   s_bi[256];
    __shared__ int   s_last;
    if (tid == 0) { fps[b*NPOINT] = 0; s_last = 0; }
    __syncthreads();
    for (int it = 1; it < NPOINT; ++it) {
        int last = s_last;
        float lx = P[last*3], ly = P[last*3+1], lz = P[last*3+2];
        float bv = -1.f; int bi = 0;
        for (int t = 0; t < 16; ++t) {
            float dx = px[t]-lx, dy = py[t]-ly, dz = pz[t]-lz;
            float d = dx*dx + dy*dy + dz*dz;
            md[t] = fminf(md[t], d);
            int i = tid + t * 256;
            if (md[t] > bv || (md[t] == bv && i < bi)) { bv = md[t]; bi = i; }
        }
        s_bv[tid] = bv; s_bi[tid] = bi;
        __syncthreads();
        for (int s = 128; s > 0; s >>= 1) {
            if (tid < s) {
                float ov = s_bv[tid+s]; int oi = s_bi[tid+s];
                if (ov > s_bv[tid] || (ov == s_bv[tid] && oi < s_bi[tid])) {
                    s_bv[tid] = ov; s_bi[tid] = oi;
                }
            }
            __syncthreads();
        }
        if (tid == 0) { fps[b*NPOINT + it] = s_bi[0]; s_last = s_bi[0]; }
        __syncthreads();
    }
}

// Ball query: one wave32 per (pct, batch, center). Ordered index compaction
// via ballot+popc exactly reproduces "nsample smallest indices within radius".
__global__ void ball_kernel(const float* __restrict__ pcd,
                            const int* __restrict__ fps,
                            int* __restrict__ ballIdx) {
    int g   = blockIdx.x;                 // [0, 5*8*204)
    int pi  = g / (BATCH * NPOINT);
    int rem = g % (BATCH * NPOINT);
    int b   = rem / NPOINT;
    int c   = rem % NPOINT;
    int lane = threadIdx.x;
    const float* P = pcd + b * NPTS * 3;
    int cidx = fps[b*NPOINT + c];
    float cx = P[cidx*3], cy = P[cidx*3+1], cz = P[cidx*3+2];
    float r2 = c_R2[pi];
    int ns   = c_NS[pi];
    int* out = ballIdx + g * NSMAX;
    int count = 0;
    for (int base = 0; base < NPTS && count < ns; base += 32) {
        int j = base + lane;
        float dx = P[j*3]-cx, dy = P[j*3+1]-cy, dz = P[j*3+2]-cz;
        float sq = dx*dx + dy*dy + dz*dz;
        bool pred = sq < r2;
        unsigned mask = __builtin_amdgcn_ballot_w32(pred);
        int rank = __popc(mask & ((1u << lane) - 1u));
        int pos = count + rank;
        if (pred && pos < ns) out[pos] = j;
        count += __popc(mask);
    }
    if (count > ns) count = ns;
    int first = out[0];   // center always in-radius -> valid
    for (int k = count + lane; k < ns; k += 32) out[k] = first;
}

// Per-group NN-distance uniformity term: one 64-thread block per group.
__global__ void group_kernel(const float* __restrict__ pcd,
                             const int* __restrict__ ballIdx,
                             float* __restrict__ out) {
    int g   = blockIdx.x;
    int pi  = g / (BATCH * NPOINT);
    int rem = g % (BATCH * NPOINT);
    int b   = rem / NPOINT;
    int tid = threadIdx.x;
    int ns  = c_NS[pi];
    const float* P = pcd + b * NPTS * 3;
    const int* idx = ballIdx + g * NSMAX;
    __shared__ float sx[NSMAX], sy[NSMAX], sz[NSMAX];
    __shared__ float red[64];
    if (tid < ns) {
        int j = idx[tid];
        sx[tid] = P[j*3]; sy[tid] = P[j*3+1]; sz[tid] = P[j*3+2];
    }
    __syncthreads();
    float ud = 0.f;
    if (tid < ns) {
        float x = sx[tid], y = sy[tid], z = sz[tid];
        float m = 1e30f;
        for (int j = 0; j < ns; ++j) {
            float dx = sx[j]-x, dy = sy[j]-y, dz = sz[j]-z;
            float d = dx*dx + dy*dy + dz*dz;
            d = (j == tid) ? 1e30f : d;   // branchless self-exclusion
            m = fminf(m, d);
        }
        float dist = sqrtf(fmaxf(m, 0.f) + 1e-12f);
        ud = fabsf(dist + 1e-8f);
    }
    red[tid] = ud;
    __syncthreads();
    for (int s = 32; s > 0; s >>= 1) {
        if (tid < s) red[tid] += red[tid + s];
        __syncthreads();
    }
    if (tid == 0) {
        float expect = sqrtf(3.14159265358979f / (float)NPTS);
        float u = red[0] / (float)ns;
        float v = (u - expect) * (u - expect) / (expect + 1e-8f);
        atomicAdd(out, v * c_W[pi]);
    }
}

// Branchless sorted insert of v into ascending {t0..t4}: pure min/max network,
// maps to v_dual_min/max_num_f32 with no exec-mask manipulation.
__device__ __forceinline__ void ins5(float v, float& t0, float& t1, float& t2,
                                     float& t3, float& t4) {
    float n4 = fminf(t4, fmaxf(t3, v));
    float n3 = fminf(t3, fmaxf(t2, v));
    float n2 = fminf(t2, fmaxf(t1, v));
    float n1 = fminf(t1, fmaxf(t0, v));
    float n0 = fminf(t0, v);
    t0 = n0; t1 = n1; t2 = n2; t3 = n3; t4 = n4;
}

#define TSTRIDE 33   // LDS row stride (floats), padded to avoid bank conflicts

// Repulsion: one wave32 per (batch, 16-row strip). Two f32 WMMA 16x16x4 Gram
// tiles per iteration (32 columns); sqd = |xi|^2+|xj|^2-2G; fused running
// top-5 per row via LDS transpose + branchless insertion.
__global__ void repulsion_kernel(const float* __restrict__ pcd,
                                 const float* __restrict__ rr,
                                 float* __restrict__ out) {
    int b     = blockIdx.x >> 8;      // / 256 strips
    int strip = blockIdx.x & 255;
    int i0    = strip * 16;
    int lane  = threadIdx.x;
    int half  = lane >> 4;
    int l15   = lane & 15;
    const float* P = pcd + b * NPTS * 3;
    const float* R = rr + b * NPTS;

    // A (16x4, MxK): lanes 0-15 hold K=0,1 (x,y); lanes 16-31 hold K=2,3 (z,0)
    // Branchless: x/z via computed component offset, y zeroed for upper half.
    {
        int ja = (i0 + l15) * 3;
        float ax = P[ja + (half << 1)];     // x (half 0) or z (half 1)
        float ay = P[ja + 1];
        ay = half ? 0.f : ay;
        v2f Atmp; Atmp.x = ax; Atmp.y = ay;
        // keep A in a named var below
        #define A_OP Atmp
        float rq[8];
        for (int k = 0; k < 8; ++k) rq[k] = R[i0 + k + half*8];

        float t0 = 3.4e38f, t1 = 3.4e38f, t2 = 3.4e38f, t3 = 3.4e38f, t4 = 3.4e38f;
        __shared__ float tile[16 * TSTRIDE];

        for (int c0 = 0; c0 < NPTS; c0 += 32) {
            int j0 = (c0 + l15) * 3;
            int j1 = (c0 + 16 + l15) * 3;
            float b0x = P[j0 + (half << 1)];
            float b0y = P[j0 + 1];
            float b1x = P[j1 + (half << 1)];
            float b1y = P[j1 + 1];
            b0y = half ? 0.f : b0y;
            b1y = half ? 0.f : b1y;
            v2f B0; B0.x = b0x; B0.y = b0y;
            v2f B1; B1.x = b1x; B1.y = b1y;

            v8f C0 = {0.f,0.f,0.f,0.f,0.f,0.f,0.f,0.f};
            v8f C1 = {0.f,0.f,0.f,0.f,0.f,0.f,0.f,0.f};
            C0 = __builtin_amdgcn_wmma_f32_16x16x4_f32(
                    false, A_OP, false, B0, (short)0, C0, false, false);
            C1 = __builtin_amdgcn_wmma_f32_16x16x4_f32(
                    false, A_OP, false, B1, (short)0, C1, false, false);

            float rc0 = R[c0 + l15];
            float rc1 = R[c0 + 16 + l15];
            for (int k = 0; k < 8; ++k) {
                int row = k + half*8;
                float s0 = fmaxf(rq[k] + rc0 - 2.f * C0[k], 0.f);
                float s1 = fmaxf(rq[k] + rc1 - 2.f * C1[k], 0.f);
                tile[row * TSTRIDE + l15]      = s0;
                tile[row * TSTRIDE + 16 + l15] = s1;
            }
            __syncthreads();
            // lane owns row l15, column segment [half*16, half*16+16)
            for (int q = 0; q < 16; ++q) {
                float v = tile[l15 * TSTRIDE + half*16 + q];
                ins5(v, t0, t1, t2, t3, t4);
            }
            __syncthreads();
        }

        __shared__ float fin[160];
        fin[lane*5+0] = t0; fin[lane*5+1] = t1; fin[lane*5+2] = t2;
        fin[lane*5+3] = t3; fin[lane*5+4] = t4;
        __syncthreads();
        if (lane < 16) {
            float vals[10];
            for (int s = 0; s < 5; ++s) {
                vals[s]     = fin[lane*5 + s];
                vals[5 + s] = fin[(lane + 16)*5 + s];
            }
            float contrib = 0.f;
            for (int sel = 0; sel < 5; ++sel) {
                int bi = 0; float bv = vals[0];
                for (int t = 1; t < 10; ++t) if (vals[t] < bv) { bv = vals[t]; bi = t; }
                vals[bi] = 3.4e38f;
                if (sel >= 1) contrib += fmaxf(0.0005f - bv, 0.f);  // entries 1..4
            }
            atomicAdd(out, contrib * 7.62939453125e-6f);  // 1/(8*4096*4)
        }
        #undef A_OP
    }
}

extern "C" void kernel_launch(void* const* d_in, const int* in_sizes, int n_in,
                              void* d_out, int out_size, void* d_ws, size_t ws_size,
                              hipStream_t stream) {
    const float* pcd = (const float*)d_in[0];
    float* out = (float*)d_out;
    char* ws = (char*)d_ws;
    float* rr     = (float*)ws;                                   // 8*4096 f32
    int*   fps    = (int*)(ws + BATCH*NPTS*sizeof(float));        // 8*204 i32
    int*   ballIx = (int*)(ws + BATCH*NPTS*sizeof(float)
                              + BATCH*NPOINT*sizeof(int));        // 5*8*204*49 i32

    zero_out_kernel<<<1, 32, 0, stream>>>(out);
    rr_kernel<<<(BATCH*NPTS + 255)/256, 256, 0, stream>>>(pcd, rr);
    fps_kernel<<<BATCH, 256, 0, stream>>>(pcd, fps);
    ball_kernel<<<NPCT*BATCH*NPOINT, 32, 0, stream>>>(pcd, fps, ballIx);
    group_kernel<<<NPCT*BATCH*NPOINT, 64, 0, stream>>>(pcd, ballIx, out);
    repulsion_kernel<<<BATCH*256, 32, 0, stream>>>(pcd, rr, out);
}